// SequenceTaggerWithBahdanauAttention_15444702397228
// MI455X (gfx1250) — compile-verified
//
#include <hip/hip_runtime.h>
#include <math.h>

// Problem dims
#define Bb 32
#define Ss 512
#define Ii 768
#define Hh 512
#define Cc 10
#define G4H (4 * Hh)   // 2048
#define I2  (2 * Ii)   // 1536

typedef float v2f __attribute__((ext_vector_type(2)));
typedef float v8f __attribute__((ext_vector_type(8)));

__device__ __forceinline__ float sigmoidf(float x) {
    return 1.0f / (1.0f + expf(-x));
}

// ---------------------------------------------------------------------------
// enc_proj = embeds(16384,768) @ Wenc(512,768)^T + benc  -> (16384, 512)
// WMMA f32 16x16x4. Each wave: 16 rows x 64 cols strip (4 accumulators).
// grid: (16384/(16*8), 512/64) = (128, 8), block: 256 (8 waves)
// ---------------------------------------------------------------------------
__global__ void encproj_kernel(const float* __restrict__ X,
                               const float* __restrict__ W,
                               const float* __restrict__ bias,
                               float* __restrict__ out) {
    const int lane = threadIdx.x & 31;
    const int wave = threadIdx.x >> 5;
    const int m0 = (blockIdx.x * 8 + wave) * 16;
    const int n0 = blockIdx.y * 64;
    const int lm = lane & 15;
    const int lk = (lane >> 4) << 1;   // 0 or 2

    v8f acc0 = {}, acc1 = {}, acc2 = {}, acc3 = {};
    const float* arow = X + (size_t)(m0 + lm) * Ii;
    const float* brow = W + (size_t)(n0 + lm) * Ii;

    for (int k = 0; k < Ii; k += 4) {
        v2f a  = *(const v2f*)(arow + k + lk);
        v2f b0 = *(const v2f*)(brow + (size_t)0 * 16 * Ii + k + lk);
        v2f b1 = *(const v2f*)(brow + (size_t)1 * 16 * Ii + k + lk);
        v2f b2 = *(const v2f*)(brow + (size_t)2 * 16 * Ii + k + lk);
        v2f b3 = *(const v2f*)(brow + (size_t)3 * 16 * Ii + k + lk);
        acc0 = __builtin_amdgcn_wmma_f32_16x16x4_f32(false, a, false, b0, (short)0, acc0, false, false);
        acc1 = __builtin_amdgcn_wmma_f32_16x16x4_f32(false, a, false, b1, (short)0, acc1, false, false);
        acc2 = __builtin_amdgcn_wmma_f32_16x16x4_f32(false, a, false, b2, (short)0, acc2, false, false);
        acc3 = __builtin_amdgcn_wmma_f32_16x16x4_f32(false, a, false, b3, (short)0, acc3, false, false);
    }

    const int rbase = m0 + ((lane >> 4) << 3);  // +8 for high half-lanes
    for (int v = 0; v < 8; ++v) {
        size_t r = (size_t)(rbase + v) * Hh;
        out[r + n0 +  0 + lm] = acc0[v] + bias[n0 +  0 + lm];
        out[r + n0 + 16 + lm] = acc1[v] + bias[n0 + 16 + lm];
        out[r + n0 + 32 + lm] = acc2[v] + bias[n0 + 32 + lm];
        out[r + n0 + 48 + lm] = acc3[v] + bias[n0 + 48 + lm];
    }
}

// ---------------------------------------------------------------------------
// Zero h/c state (2 dirs x 32 x 512 each)
// ---------------------------------------------------------------------------
__global__ void init_state_kernel(float* __restrict__ h, float* __restrict__ c) {
    int idx = blockIdx.x * blockDim.x + threadIdx.x;
    if (idx < 2 * Bb * Hh) { h[idx] = 0.0f; c[idx] = 0.0f; }
}

// ---------------------------------------------------------------------------
// Attention for one timestep. One block per (b, dir): 512 threads.
// dec = h@Wdec^T+bdec ; scores = v . tanh(enc_proj + dec) ; softmax ; ctx
// ---------------------------------------------------------------------------
__global__ void attn_kernel(int t,
                            const float* __restrict__ embeds,
                            const float* __restrict__ encp,
                            const float* __restrict__ Wdec,
                            const float* __restrict__ bdec,
                            const float* __restrict__ vvec,
                            const float* __restrict__ h_state,
                            float* __restrict__ ctx_out) {
    __shared__ float sh_h[Hh];
    __shared__ float sh_dec[Hh];
    __shared__ float sh_w[Ss];
    __shared__ float sh_r[Ss];

    const int b = blockIdx.x;
    const int d = blockIdx.y;
    const int tid = threadIdx.x;

    sh_h[tid] = h_state[(d * Bb + b) * Hh + tid];
    __syncthreads();

    // dec[tid]
    {
        float acc = bdec[tid];
        const float* wr = Wdec + (size_t)tid * Hh;
        for (int k = 0; k < Hh; ++k) acc += sh_h[k] * wr[k];
        sh_dec[tid] = acc;
    }
    __syncthreads();

    // score for s = tid
    float sc = 0.0f;
    {
        const float* ep = encp + ((size_t)b * Ss + tid) * Hh;
        for (int k = 0; k < Hh; ++k) sc += vvec[k] * tanhf(ep[k] + sh_dec[k]);
    }
    sh_r[tid] = sc;
    __syncthreads();
    for (int off = 256; off > 0; off >>= 1) {
        if (tid < off) sh_r[tid] = fmaxf(sh_r[tid], sh_r[tid + off]);
        __syncthreads();
    }
    float mx = sh_r[0];
    __syncthreads();
    float e = expf(sc - mx);
    sh_r[tid] = e;
    __syncthreads();
    for (int off = 256; off > 0; off >>= 1) {
        if (tid < off) sh_r[tid] += sh_r[tid + off];
        __syncthreads();
    }
    sh_w[tid] = e / sh_r[0];
    __syncthreads();

    // ctx[i] = sum_s w[s] * embeds[b, (d? S-1-s : s), i]
    for (int i = tid; i < Ii; i += Ss) {
        float acc = 0.0f;
        for (int s = 0; s < Ss; ++s) {
            int sr = d ? (Ss - 1 - s) : s;
            acc += sh_w[s] * embeds[((size_t)b * Ss + sr) * Ii + i];
        }
        ctx_out[(d * Bb + b) * Ii + i] = acc;
    }
}

// ---------------------------------------------------------------------------
// Gates GEMM: gates[d] = [ctx | x_t](32,1536) @ Wih^T + h(32,512) @ Whh^T + b
// WMMA f32 16x16x4, one 16x16 tile per wave.
// tiles: 2 dir x 2 Mtiles x 128 Ntiles = 512 waves -> grid 64 x 256 threads
// ---------------------------------------------------------------------------
__global__ void gates_kernel(int t,
                             const float* __restrict__ embeds,
                             const float* __restrict__ ctx,
                             const float* __restrict__ h_state,
                             const float* __restrict__ Wih_f,
                             const float* __restrict__ Whh_f,
                             const float* __restrict__ bih_f,
                             const float* __restrict__ bhh_f,
                             const float* __restrict__ Wih_b,
                             const float* __restrict__ Whh_b,
                             const float* __restrict__ bih_b,
                             const float* __restrict__ bhh_b,
                             float* __restrict__ gates) {
    const int lane = threadIdx.x & 31;
    const int wave = threadIdx.x >> 5;
    const int tile = blockIdx.x * 8 + wave;      // 0..511
    const int d  = tile >> 8;                    // 0..1
    const int mt = (tile >> 7) & 1;              // 0..1
    const int nt = tile & 127;                   // 0..127
    const int m0 = mt * 16;
    const int n0 = nt * 16;
    const int lm = lane & 15;
    const int lk = (lane >> 4) << 1;

    const float* Wih = d ? Wih_b : Wih_f;
    const float* Whh = d ? Whh_b : Whh_f;
    const float* bih = d ? bih_b : bih_f;
    const float* bhh = d ? bhh_b : bhh_f;

    const int   brow  = m0 + lm;                        // batch row 0..31
    const int   tstep = d ? (Ss - 1 - t) : t;
    const float* actx = ctx + (d * Bb + brow) * Ii;
    const float* axt  = embeds + ((size_t)brow * Ss + tstep) * Ii;
    const float* ah   = h_state + (d * Bb + brow) * Hh;
    const float* bwih = Wih + (size_t)(n0 + lm) * I2;
    const float* bwhh = Whh + (size_t)(n0 + lm) * Hh;

    v8f acc = {};
    for (int k = 0; k < Ii; k += 4) {                   // ctx part (K 0..767)
        v2f a = *(const v2f*)(actx + k + lk);
        v2f b = *(const v2f*)(bwih + k + lk);
        acc = __builtin_amdgcn_wmma_f32_16x16x4_f32(false, a, false, b, (short)0, acc, false, false);
    }
    for (int k = 0; k < Ii; k += 4) {                   // x_t part (K 768..1535)
        v2f a = *(const v2f*)(axt + k + lk);
        v2f b = *(const v2f*)(bwih + Ii + k + lk);
        acc = __builtin_amdgcn_wmma_f32_16x16x4_f32(false, a, false, b, (short)0, acc, false, false);
    }
    for (int k = 0; k < Hh; k += 4) {                   // h part
        v2f a = *(const v2f*)(ah + k + lk);
        v2f b = *(const v2f*)(bwhh + k + lk);
        acc = __builtin_amdgcn_wmma_f32_16x16x4_f32(false, a, false, b, (short)0, acc, false, false);
    }

    const int rbase = m0 + ((lane >> 4) << 3);
    const float bias = bih[n0 + lm] + bhh[n0 + lm];
    for (int v = 0; v < 8; ++v) {
        gates[((size_t)d * Bb + (rbase + v)) * G4H + n0 + lm] = acc[v] + bias;
    }
}

// ---------------------------------------------------------------------------
// LSTM cell update + write hidden state into combined hs buffer.
// idx over 2 x 32 x 512
// ---------------------------------------------------------------------------
__global__ void cell_kernel(int t,
                            const float* __restrict__ gates,
                            float* __restrict__ h_state,
                            float* __restrict__ c_state,
                            float* __restrict__ hs) {
    int idx = blockIdx.x * blockDim.x + threadIdx.x;
    if (idx >= 2 * Bb * Hh) return;
    int d = idx >> 14;
    int b = (idx >> 9) & (Bb - 1);
    int hh = idx & (Hh - 1);
    const float* g = gates + ((size_t)d * Bb + b) * G4H;
    float gi = g[hh], gf = g[Hh + hh], gg = g[2 * Hh + hh], go = g[3 * Hh + hh];
    float c  = c_state[idx];
    float cn = sigmoidf(gf) * c + sigmoidf(gi) * tanhf(gg);
    float hn = sigmoidf(go) * tanhf(cn);
    c_state[idx] = cn;
    h_state[idx] = hn;
    int tout = d ? (Ss - 1 - t) : t;
    hs[((size_t)tout * Bb + b) * (2 * Hh) + d * Hh + hh] = hn;
}

// ---------------------------------------------------------------------------
// Output projection: out[b,t,c] = combined[t,b,:] . Wout[c,:] + bout[c]
// ---------------------------------------------------------------------------
__global__ void outproj_kernel(const float* __restrict__ hs,
                               const float* __restrict__ Wout,
                               const float* __restrict__ bout,
                               float* __restrict__ out) {
    int idx = blockIdx.x * blockDim.x + threadIdx.x;
    if (idx >= Bb * Ss * Cc) return;
    int c = idx % Cc;
    int t = (idx / Cc) % Ss;
    int b = idx / (Cc * Ss);
    const float* row = hs + ((size_t)t * Bb + b) * (2 * Hh);
    const float* wr  = Wout + (size_t)c * (2 * Hh);
    float acc = bout[c];
    for (int j = 0; j < 2 * Hh; ++j) acc += row[j] * wr[j];
    out[idx] = acc;
}

// ---------------------------------------------------------------------------
extern "C" void kernel_launch(void* const* d_in, const int* in_sizes, int n_in,
                              void* d_out, int out_size, void* d_ws, size_t ws_size,
                              hipStream_t stream) {
    const float* embeds = (const float*)d_in[0];
    const float* Wenc   = (const float*)d_in[1];
    const float* benc   = (const float*)d_in[2];
    const float* Wdec   = (const float*)d_in[3];
    const float* bdec   = (const float*)d_in[4];
    const float* vvec   = (const float*)d_in[5];
    const float* Wih_f  = (const float*)d_in[6];
    const float* Whh_f  = (const float*)d_in[7];
    const float* bih_f  = (const float*)d_in[8];
    const float* bhh_f  = (const float*)d_in[9];
    const float* Wih_b  = (const float*)d_in[10];
    const float* Whh_b  = (const float*)d_in[11];
    const float* bih_b  = (const float*)d_in[12];
    const float* bhh_b  = (const float*)d_in[13];
    const float* Wout   = (const float*)d_in[14];
    const float* bout   = (const float*)d_in[15];
    float* out = (float*)d_out;

    // Workspace layout (floats), ~102 MB total
    float* ws    = (float*)d_ws;
    float* encp  = ws;                                   // 32*512*512   = 8388608
    float* ctx   = encp  + (size_t)Bb * Ss * Hh;         // 2*32*768     = 49152
    float* gates = ctx   + (size_t)2 * Bb * Ii;          // 2*32*2048    = 131072
    float* hst   = gates + (size_t)2 * Bb * G4H;         // 2*32*512     = 32768
    float* cst   = hst   + (size_t)2 * Bb * Hh;          // 2*32*512     = 32768
    float* hs    = cst   + (size_t)2 * Bb * Hh;          // 512*32*1024  = 16777216

    // 1) enc_proj (one-shot WMMA GEMM)
    encproj_kernel<<<dim3(128, 8), 256, 0, stream>>>(embeds, Wenc, benc, encp);

    // 2) zero recurrent state
    init_state_kernel<<<(2 * Bb * Hh + 255) / 256, 256, 0, stream>>>(hst, cst);

    // 3) sequential scan, both directions fused per step
    for (int t = 0; t < Ss; ++t) {
        attn_kernel<<<dim3(Bb, 2), Ss, 0, stream>>>(t, embeds, encp, Wdec, bdec, vvec, hst, ctx);
        gates_kernel<<<64, 256, 0, stream>>>(t, embeds, ctx, hst,
                                             Wih_f, Whh_f, bih_f, bhh_f,
                                             Wih_b, Whh_b, bih_b, bhh_b, gates);
        cell_kernel<<<(2 * Bb * Hh + 255) / 256, 256, 0, stream>>>(t, gates, hst, cst, hs);
    }

    // 4) output projection
    outproj_kernel<<<(Bb * Ss * Cc + 255) / 256, 256, 0, stream>>>(hs, Wout, bout, out);

    (void)in_sizes; (void)n_in; (void)out_size; (void)ws_size;
}